// RIPGeo_68710886801531
// MI455X (gfx1250) — compile-verified
//
#include <hip/hip_runtime.h>
#include <hip/hip_bf16.h>

typedef float v2f __attribute__((ext_vector_type(2)));
typedef float v8f __attribute__((ext_vector_type(8)));

#define N1 2000
#define N2 4000
#define DIN 30
#define NT1 (N1/16)   /* 125 landmark tiles */
#define NT2 (N2/16)   /* 250 target tiles  */
#define INV_TEMP 0.17677669529663687f  /* 1/sqrt(32) */

/* workspace layout (float offsets) */
#define WS_KMAT   0
#define WS_PKMAT  (WS_KMAT  + N1*32)
#define WS_PVMAT  (WS_PKMAT + N1*32)
#define WS_LMF    (WS_PVMAT + N1*2)
#define WS_DELAY  (WS_LMF   + N1*32)
#define WS_QS     (WS_DELAY + N1)
#define WS_ROU0   (WS_QS    + N2*32)
#define WS_ROU1   (WS_ROU0  + N2)
#define WS_R0     (WS_ROU1  + N2)
#define WS_WSUM   (WS_R0    + 32)
#define WS_SD     (WS_WSUM  + 32)
#define WS_R1     (WS_SD    + 32)
#define WS_UMAT   (WS_R1    + 32)
#define WS_UMAT2  (WS_UMAT  + N2*32)
#define WS_PQS    (WS_UMAT2 + N2*32)

__device__ __forceinline__ v8f wmma4(v2f a, v2f b, v8f c) {
    return __builtin_amdgcn_wmma_f32_16x16x4_f32(false, a, false, b, (short)0, c, false, false);
}

/* ---- landmark-side prep: k, pk, pv, lm_feature, delay_score ---- */
__global__ void prep_lm(const float* lm_X, const float* lm_Y, const float* lm_delay,
                        const float* g1, const float* alpha, const float* beta,
                        const float* att_kw, const float* att_kb,
                        const float* pred_kw, const float* pred_kb,
                        const float* pred_vw, const float* pred_vb, float* ws) {
    int i = blockIdx.x * blockDim.x + threadIdx.x;
    if (i >= N1) return;
    float x[DIN];
    for (int c = 0; c < DIN; ++c) x[c] = lm_X[i*DIN + c];
    float y0 = lm_Y[i*2], y1 = lm_Y[i*2 + 1];
    float* lmf = ws + WS_LMF + i*32;
    for (int c = 0; c < DIN; ++c) lmf[c] = x[c];
    lmf[30] = y0; lmf[31] = y1;
    float* km  = ws + WS_KMAT  + i*32;
    float* pkm = ws + WS_PKMAT + i*32;
    for (int z = 0; z < 32; ++z) {
        float a1 = att_kb[z], a2 = pred_kb[z];
        for (int c = 0; c < DIN; ++c) {
            a1 += x[c] * att_kw[c*32 + z];
            a2 += x[c] * pred_kw[c*32 + z];
        }
        km[z] = a1; pkm[z] = a2;
    }
    ws[WS_PVMAT + i*2 + 0] = y0*pred_vw[0] + y1*pred_vw[2] + pred_vb[0];
    ws[WS_PVMAT + i*2 + 1] = y0*pred_vw[1] + y1*pred_vw[3] + pred_vb[1];
    ws[WS_DELAY + i] = __expf(-g1[0] * (alpha[0]*lm_delay[i] + beta[0]));
}

/* ---- column reductions: router0 (mean), sum(delay*lmf), sum(delay) ---- */
__global__ void reduce_lm(float* ws) {
    __shared__ float s0[256], s1[256], s2[256];
    int j = blockIdx.x, t = threadIdx.x;
    float a0 = 0.f, a1 = 0.f, a2 = 0.f;
    for (int i = t; i < N1; i += 256) {
        float f = ws[WS_LMF + i*32 + j];
        float d = ws[WS_DELAY + i];
        a0 += f; a1 += d*f; a2 += d;
    }
    s0[t] = a0; s1[t] = a1; s2[t] = a2;
    __syncthreads();
    for (int o = 128; o > 0; o >>= 1) {
        if (t < o) { s0[t] += s0[t+o]; s1[t] += s1[t+o]; s2[t] += s2[t+o]; }
        __syncthreads();
    }
    if (t == 0) {
        ws[WS_R0 + j]   = s0[0] / (float)N1;
        ws[WS_WSUM + j] = s1[0];
        if (j == 0) ws[WS_SD] = s2[0];
    }
}

/* ---- target-side prep: q (pre-scaled), rou2tar_0/1, tg_X into final feature ---- */
__global__ void prep_tg(const float* tg_X, const float* tg_delay,
                        const float* g2, const float* g3, const float* alpha, const float* beta,
                        const float* att_qw, const float* att_qb, float* ws, float* out) {
    int i = blockIdx.x * blockDim.x + threadIdx.x;
    if (i >= N2) return;
    float x[DIN];
    for (int c = 0; c < DIN; ++c) x[c] = tg_X[i*DIN + c];
    float* fo = out + 2*N2 + i*94;
    for (int c = 0; c < DIN; ++c) fo[c] = x[c];
    float* q = ws + WS_QS + i*32;
    for (int z = 0; z < 32; ++z) {
        float a = att_qb[z];
        for (int c = 0; c < DIN; ++c) a += x[c] * att_qw[c*32 + z];
        q[z] = a * INV_TEMP;
    }
    float d = tg_delay[i];
    ws[WS_ROU0 + i] = __expf(-g2[0] * (alpha[0]*d + beta[0]));
    ws[WS_ROU1 + i] = __expf(-g3[0] * (alpha[0]*d + beta[0]));
}

/* ---- router_1 = ((wsum + router0)/(1+sum_delay)) @ w1 + b1 ---- */
__global__ void router1_k(const float* w1_w, const float* w1_b, float* ws) {
    int z = threadIdx.x;                  /* 32 threads */
    float inv = 1.0f / (1.0f + ws[WS_SD]);
    float acc = w1_b[z];
    for (int c = 0; c < 32; ++c) {
        float rb = (ws[WS_WSUM + c] + ws[WS_R0 + c]) * inv;
        acc += rb * w1_w[c*32 + z];
    }
    ws[WS_R1 + z] = acc;
}

/* ---- attention 1 (flash, fp32 WMMA): u = (softmax(qk')@lmf + tgf0 + r0*router0)/(2+r0) ---- */
__global__ void __launch_bounds__(256) attn1(const float* tg_X, float* ws) {
    const int lane = threadIdx.x & 31;
    const int wv   = threadIdx.x >> 5;
    const int tile = blockIdx.x * 8 + wv;
    if (tile >= NT2) return;
    const int half = lane >> 4, l16 = lane & 15;
    __shared__ float sP[8][256];
    const float* qs  = ws + WS_QS;
    const float* km  = ws + WS_KMAT;
    const float* lmf = ws + WS_LMF;

    v2f a[8];
    {
        const float* qrow = qs + (tile*16 + l16)*32 + 2*half;
        for (int c = 0; c < 8; ++c) a[c] = *(const v2f*)(qrow + 4*c);
    }
    /* pass 1: lane-local max, one cross-lane reduction at the end */
    float m[8];
    for (int v = 0; v < 8; ++v) m[v] = -3.0e38f;
    for (int j = 0; j < NT1; ++j) {
        const float* krow = km + (j*16 + l16)*32 + 2*half;
        v8f s = {0.f,0.f,0.f,0.f,0.f,0.f,0.f,0.f};
        for (int c = 0; c < 8; ++c) s = wmma4(a[c], *(const v2f*)(krow + 4*c), s);
        for (int v = 0; v < 8; ++v) m[v] = fmaxf(m[v], s[v]);
    }
    for (int v = 0; v < 8; ++v)
        for (int o = 1; o < 16; o <<= 1) m[v] = fmaxf(m[v], __shfl_xor(m[v], o, 16));
    /* pass 2: P = exp(S-m), l += rowsum, O += P@V (via LDS layout swap) */
    float l[8];
    for (int v = 0; v < 8; ++v) l[v] = 0.f;
    v8f o0 = {0.f,0.f,0.f,0.f,0.f,0.f,0.f,0.f};
    v8f o1 = {0.f,0.f,0.f,0.f,0.f,0.f,0.f,0.f};
    for (int j = 0; j < NT1; ++j) {
        const float* krow = km + (j*16 + l16)*32 + 2*half;
        v8f s = {0.f,0.f,0.f,0.f,0.f,0.f,0.f,0.f};
        for (int c = 0; c < 8; ++c) s = wmma4(a[c], *(const v2f*)(krow + 4*c), s);
        for (int v = 0; v < 8; ++v) {
            float p = __expf(s[v] - m[v]);
            l[v] += p;
            sP[wv][(v + 8*half)*16 + l16] = p;   /* C/D layout -> row-major tile */
        }
        asm volatile("s_wait_dscnt 0x0" ::: "memory");
        const float* vb = lmf + j*16*32;
        for (int c = 0; c < 4; ++c) {
            int kk = 4*c + 2*half;
            v2f pa; pa.x = sP[wv][l16*16 + kk]; pa.y = sP[wv][l16*16 + kk + 1];
            v2f b0, b1;
            b0.x = vb[kk*32 + l16];      b0.y = vb[(kk+1)*32 + l16];
            b1.x = vb[kk*32 + 16 + l16]; b1.y = vb[(kk+1)*32 + 16 + l16];
            o0 = wmma4(pa, b0, o0);
            o1 = wmma4(pa, b1, o1);
        }
        asm volatile("s_wait_dscnt 0x0" ::: "memory");
    }
    for (int v = 0; v < 8; ++v)
        for (int o = 1; o < 16; o <<= 1) l[v] += __shfl_xor(l[v], o, 16);
    /* epilogue: fuse adj0/deg0 normalization */
    const float* r0v = ws + WS_R0;
    for (int v = 0; v < 8; ++v) {
        int i = tile*16 + v + 8*half;
        float r0 = ws[WS_ROU0 + i];
        float invl = 1.0f / l[v];
        float invdeg = 1.0f / (2.0f + r0);
        int n0 = l16, n1c = 16 + l16;
        float t0 = (n0  < DIN) ? tg_X[i*DIN + n0]  : 0.f;
        float t1 = (n1c < DIN) ? tg_X[i*DIN + n1c] : 0.f;
        ws[WS_UMAT + i*32 + n0]  = (o0[v]*invl + t0 + r0*r0v[n0])  * invdeg;
        ws[WS_UMAT + i*32 + n1c] = (o1[v]*invl + t1 + r0*r0v[n1c]) * invdeg;
    }
}

/* ---- tf1 = u@w1+b1 -> out cols 30..61 ; u2 = (tf1 + r1*router1)/(1+r1) ---- */
__global__ void gemm_w1(const float* w1_w, const float* w1_b, float* ws, float* out) {
    int idx = blockIdx.x * blockDim.x + threadIdx.x;
    if (idx >= N2*32) return;
    int i = idx >> 5, z = idx & 31;
    const float* u = ws + WS_UMAT + i*32;
    float acc = w1_b[z];
    for (int c = 0; c < 32; ++c) acc += u[c] * w1_w[c*32 + z];
    out[2*N2 + i*94 + 30 + z] = acc;
    float r1 = ws[WS_ROU1 + i];
    ws[WS_UMAT2 + i*32 + z] = (acc + r1 * ws[WS_R1 + z]) / (1.0f + r1);
}

/* ---- tf2 = u2@w2+b2 -> out cols 62..93 ---- */
__global__ void gemm_w2(const float* w2_w, const float* w2_b, float* ws, float* out) {
    int idx = blockIdx.x * blockDim.x + threadIdx.x;
    if (idx >= N2*32) return;
    int i = idx >> 5, z = idx & 31;
    const float* u = ws + WS_UMAT2 + i*32;
    float acc = w2_b[z];
    for (int c = 0; c < 32; ++c) acc += u[c] * w2_w[c*32 + z];
    out[2*N2 + i*94 + 62 + z] = acc;
}

/* ---- pq = final_feature(94) @ pred_qw + b, pre-scaled by 1/TEMP ---- */
__global__ void pq_k(const float* pred_qw, const float* pred_qb, const float* out, float* ws) {
    int idx = blockIdx.x * blockDim.x + threadIdx.x;
    if (idx >= N2*32) return;
    int i = idx >> 5, z = idx & 31;
    const float* f = out + 2*N2 + i*94;
    float acc = pred_qb[z];
    for (int c = 0; c < 94; ++c) acc += f[c] * pred_qw[c*32 + z];
    ws[WS_PQS + i*32 + z] = acc * INV_TEMP;
}

/* ---- attention 2: y = softmax(pq pk')@pv (QK^T via WMMA, P@V lane-local) ---- */
__global__ void __launch_bounds__(256) attn2(float* ws, float* out) {
    const int lane = threadIdx.x & 31;
    const int wv   = threadIdx.x >> 5;
    const int tile = blockIdx.x * 8 + wv;
    if (tile >= NT2) return;
    const int half = lane >> 4, l16 = lane & 15;
    const float* qs = ws + WS_PQS;
    const float* km = ws + WS_PKMAT;
    v2f a[8];
    {
        const float* qrow = qs + (tile*16 + l16)*32 + 2*half;
        for (int c = 0; c < 8; ++c) a[c] = *(const v2f*)(qrow + 4*c);
    }
    float m[8];
    for (int v = 0; v < 8; ++v) m[v] = -3.0e38f;
    for (int j = 0; j < NT1; ++j) {
        const float* krow = km + (j*16 + l16)*32 + 2*half;
        v8f s = {0.f,0.f,0.f,0.f,0.f,0.f,0.f,0.f};
        for (int c = 0; c < 8; ++c) s = wmma4(a[c], *(const v2f*)(krow + 4*c), s);
        for (int v = 0; v < 8; ++v) m[v] = fmaxf(m[v], s[v]);
    }
    for (int v = 0; v < 8; ++v)
        for (int o = 1; o < 16; o <<= 1) m[v] = fmaxf(m[v], __shfl_xor(m[v], o, 16));
    float l[8], a0[8], a1[8];
    for (int v = 0; v < 8; ++v) { l[v] = 0.f; a0[v] = 0.f; a1[v] = 0.f; }
    for (int j = 0; j < NT1; ++j) {
        const float* krow = km + (j*16 + l16)*32 + 2*half;
        v8f s = {0.f,0.f,0.f,0.f,0.f,0.f,0.f,0.f};
        for (int c = 0; c < 8; ++c) s = wmma4(a[c], *(const v2f*)(krow + 4*c), s);
        v2f pv = *(const v2f*)(ws + WS_PVMAT + (j*16 + l16)*2);
        for (int v = 0; v < 8; ++v) {
            float p = __expf(s[v] - m[v]);
            l[v]  += p;
            a0[v] += p * pv.x;
            a1[v] += p * pv.y;
        }
    }
    for (int v = 0; v < 8; ++v)
        for (int o = 1; o < 16; o <<= 1) {
            l[v]  += __shfl_xor(l[v],  o, 16);
            a0[v] += __shfl_xor(a0[v], o, 16);
            a1[v] += __shfl_xor(a1[v], o, 16);
        }
    if (l16 == 0) {
        for (int v = 0; v < 8; ++v) {
            int i = tile*16 + v + 8*half;
            out[i*2 + 0] = a0[v] / l[v];
            out[i*2 + 1] = a1[v] / l[v];
        }
    }
}

extern "C" void kernel_launch(void* const* d_in, const int* in_sizes, int n_in,
                              void* d_out, int out_size, void* d_ws, size_t ws_size,
                              hipStream_t stream) {
    (void)in_sizes; (void)n_in; (void)out_size; (void)ws_size;
    const float* lm_X     = (const float*)d_in[0];
    const float* lm_Y     = (const float*)d_in[1];
    const float* tg_X     = (const float*)d_in[2];
    const float* lm_delay = (const float*)d_in[4];
    const float* tg_delay = (const float*)d_in[5];
    const float* g1       = (const float*)d_in[6];
    const float* g2       = (const float*)d_in[7];
    const float* g3       = (const float*)d_in[8];
    const float* alpha    = (const float*)d_in[9];
    const float* beta     = (const float*)d_in[10];
    const float* att_qw   = (const float*)d_in[11];
    const float* att_qb   = (const float*)d_in[12];
    const float* att_kw   = (const float*)d_in[13];
    const float* att_kb   = (const float*)d_in[14];
    const float* w1_w     = (const float*)d_in[17];
    const float* w1_b     = (const float*)d_in[18];
    const float* w2_w     = (const float*)d_in[19];
    const float* w2_b     = (const float*)d_in[20];
    const float* pred_qw  = (const float*)d_in[21];
    const float* pred_qb  = (const float*)d_in[22];
    const float* pred_kw  = (const float*)d_in[23];
    const float* pred_kb  = (const float*)d_in[24];
    const float* pred_vw  = (const float*)d_in[25];
    const float* pred_vb  = (const float*)d_in[26];
    float* out = (float*)d_out;
    float* ws  = (float*)d_ws;

    prep_lm<<<(N1 + 63) / 64, 64, 0, stream>>>(lm_X, lm_Y, lm_delay, g1, alpha, beta,
                                               att_kw, att_kb, pred_kw, pred_kb,
                                               pred_vw, pred_vb, ws);
    reduce_lm<<<32, 256, 0, stream>>>(ws);
    prep_tg<<<(N2 + 63) / 64, 64, 0, stream>>>(tg_X, tg_delay, g2, g3, alpha, beta,
                                               att_qw, att_qb, ws, out);
    router1_k<<<1, 32, 0, stream>>>(w1_w, w1_b, ws);
    attn1<<<(NT2 + 7) / 8, 256, 0, stream>>>(tg_X, ws);
    gemm_w1<<<(N2*32 + 255) / 256, 256, 0, stream>>>(w1_w, w1_b, ws, out);
    gemm_w2<<<(N2*32 + 255) / 256, 256, 0, stream>>>(w2_w, w2_b, ws, out);
    pq_k<<<(N2*32 + 255) / 256, 256, 0, stream>>>(pred_qw, pred_qb, out, ws);
    attn2<<<(NT2 + 7) / 8, 256, 0, stream>>>(ws, out);
}